// Camera_15092515078318
// MI455X (gfx1250) — compile-verified
//
#include <hip/hip_runtime.h>
#include <hip/hip_bf16.h>

typedef float v2f __attribute__((ext_vector_type(2)));
typedef float v8f __attribute__((ext_vector_type(8)));

#define RELAXC 1.3f

// Swap half-waves: lane i <-> lane i^16 (ds_swizzle group-of-32: xor=0x10, and=0x1f)
__device__ __forceinline__ float swap16f(float x) {
    return __int_as_float(__builtin_amdgcn_ds_swizzle(__float_as_int(x), 0x401F));
}

// Hardware reciprocal (v_rcp_f32, ~1 ulp) instead of the IEEE div expansion
__device__ __forceinline__ float fast_rcp(float x) {
    return __builtin_amdgcn_rcpf(x);
}

__global__ __launch_bounds__(256) void camera_project_kernel(
    const float* __restrict__ sc,    // screen_coords [N,3]
    const float* __restrict__ pos,   // pos3d        [N,3]
    const float* __restrict__ cov,   // cov3d        [N,9]
    const float* __restrict__ w2m,   // world2model  [16]
    const float* __restrict__ proj,  // projection   [16]
    float* __restrict__ outPos,      // [N,3]
    float* __restrict__ outCov,      // [N,4]
    float* __restrict__ outMask,     // [N]
    int n)
{
    const int lane = threadIdx.x & 31;
    const bool hi = lane >= 16;

    // Uniform matrix loads (wave-uniform addresses -> scalar loads)
    float W[16], P[16];
    #pragma unroll
    for (int i = 0; i < 16; ++i) { W[i] = w2m[i]; P[i] = proj[i]; }

    // Build per-lane A fragments for V_WMMA_F32_16X16X4_F32.
    // clip = (P@W) @ homo. A layout: lanes 0-15 -> (M=lane, K=0,1),
    // lanes 16-31 -> (M=lane-16, K=2,3).  A1: MVP in rows 0..3, A2: MVP in rows 8..11.
    // MVP[r][c] = sum_j P[r][j] * W[j][c]; this lane only needs columns (c0, c0+1).
    float wa[4], wb[4];
    #pragma unroll
    for (int j = 0; j < 4; ++j) {
        wa[j] = hi ? W[4*j + 2] : W[4*j + 0];
        wb[j] = hi ? W[4*j + 3] : W[4*j + 1];
    }
    float ex[4], ey[4];
    #pragma unroll
    for (int r = 0; r < 4; ++r) {
        ex[r] = P[4*r+0]*wa[0] + P[4*r+1]*wa[1] + P[4*r+2]*wa[2] + P[4*r+3]*wa[3];
        ey[r] = P[4*r+0]*wb[0] + P[4*r+1]*wb[1] + P[4*r+2]*wb[2] + P[4*r+3]*wb[3];
    }
    const int r = lane & 15;
    v2f a1, a2;
    a1.x = (r==0)?ex[0]:(r==1)?ex[1]:(r==2)?ex[2]:(r==3)?ex[3]:0.0f;
    a1.y = (r==0)?ey[0]:(r==1)?ey[1]:(r==2)?ey[2]:(r==3)?ey[3]:0.0f;
    a2.x = (r==8)?ex[0]:(r==9)?ex[1]:(r==10)?ex[2]:(r==11)?ex[3]:0.0f;
    a2.y = (r==8)?ey[0]:(r==9)?ey[1]:(r==10)?ey[2]:(r==11)?ey[3]:0.0f;

    // Wr = world2model[:3,:3]^T (constant per dispatch)
    float Wr[3][3];
    #pragma unroll
    for (int i = 0; i < 3; ++i)
        #pragma unroll
        for (int j = 0; j < 3; ++j)
            Wr[i][j] = W[4*j + i];

    const long long nn = n;
    const long long stride = (long long)gridDim.x * blockDim.x;
    // Loop condition is wave-uniform (p - lane == wave base), so EXEC is all-ones
    // at every WMMA, as the ISA requires.
    for (long long p = (long long)blockIdx.x * blockDim.x + threadIdx.x;
         (p - lane) < nn; p += stride) {
        const bool valid = p < nn;
        const long long pc = valid ? p : (nn - 1);

        // Streaming, touch-once data: non-temporal loads
        const float px = __builtin_nontemporal_load(&pos[3*pc + 0]);
        const float py = __builtin_nontemporal_load(&pos[3*pc + 1]);
        const float pz = __builtin_nontemporal_load(&pos[3*pc + 2]);

        // B fragments (4x16): column = lane&15; K0/K1 in lanes 0-15, K2/K3 in 16-31.
        // B1 column j = homo(point base+j), B2 column j = homo(point base+16+j).
        const float ox_ = swap16f(px);
        const float oy_ = swap16f(py);
        const float oz_ = swap16f(pz);
        v2f b1, b2;
        b1.x = hi ? oz_  : px;    // K=2 (z) : K=0 (x)
        b1.y = hi ? 1.0f : py;    // K=3 (1) : K=1 (y)
        b2.x = hi ? pz   : ox_;
        b2.y = hi ? 1.0f : oy_;

        v8f d = {0.f,0.f,0.f,0.f,0.f,0.f,0.f,0.f};
        d = __builtin_amdgcn_wmma_f32_16x16x4_f32(false, a1, false, b1,
                                                  (short)0, d, false, false);
        d = __builtin_amdgcn_wmma_f32_16x16x4_f32(false, a2, false, b2,
                                                  (short)0, d, false, false);
        // D layout: VGPR v = row v (lanes 0-15) / row 8+v (lanes 16-31)
        // -> every lane holds its OWN point's clip coords in d[0..3].
        const float cx = d[0], cy = d[1], cz = d[2], cw = d[3];

        const float invw = fast_rcp(fmaxf(cw, 1e-6f));
        const float nx = cx * invw, ny = cy * invw, nz = cz * invw;
        const bool m = (nz >= 0.2f) && (nz <= RELAXC) &&
                       (fabsf(nx) <= RELAXC) && (fabsf(ny) <= RELAXC);

        const float scx = __builtin_nontemporal_load(&sc[3*pc + 0]);
        const float scy = __builtin_nontemporal_load(&sc[3*pc + 1]);
        const float scz = __builtin_nontemporal_load(&sc[3*pc + 2]);
        const float s0 = scx + nx, s1 = scy + ny, s2 = scz + nz;
        float o0 = 0.5f * (s0 + 1.0f) * 1920.0f;
        float o1 = (1.0f - 0.5f * (s1 + 1.0f)) * 1080.0f;
        float o2 = s2;

        // Jacobian uses raw world coords (reproduced faithfully from reference).
        const float iz = fast_rcp(pz);
        float jw0[3], jw1[3];
        #pragma unroll
        for (int c = 0; c < 3; ++c) {
            jw0[c] = iz * (Wr[0][c] - px * Wr[2][c]);
            jw1[c] = iz * (Wr[1][c] - py * Wr[2][c]);
        }
        float cv[9];
        #pragma unroll
        for (int i = 0; i < 9; ++i)
            cv[i] = __builtin_nontemporal_load(&cov[9*pc + i]);
        float t0[3], t1[3];
        #pragma unroll
        for (int c = 0; c < 3; ++c) {
            t0[c] = jw0[0]*cv[c] + jw0[1]*cv[3+c] + jw0[2]*cv[6+c];
            t1[c] = jw1[0]*cv[c] + jw1[1]*cv[3+c] + jw1[2]*cv[6+c];
        }
        float c00 = t0[0]*jw0[0] + t0[1]*jw0[1] + t0[2]*jw0[2];
        float c01 = t0[0]*jw1[0] + t0[1]*jw1[1] + t0[2]*jw1[2];
        float c10 = t1[0]*jw0[0] + t1[1]*jw0[1] + t1[2]*jw0[2];
        float c11 = t1[0]*jw1[0] + t1[1]*jw1[1] + t1[2]*jw1[2];

        const float mf = m ? 1.0f : 0.0f;
        if (!m) { o0 = 0.0f; o1 = 0.0f; o2 = 0.0f;
                  c00 = 0.0f; c01 = 0.0f; c10 = 0.0f; c11 = 0.0f; }

        if (valid) {
            __builtin_nontemporal_store(o0, &outPos[3*p + 0]);
            __builtin_nontemporal_store(o1, &outPos[3*p + 1]);
            __builtin_nontemporal_store(o2, &outPos[3*p + 2]);
            __builtin_nontemporal_store(c00, &outCov[4*p + 0]);
            __builtin_nontemporal_store(c01, &outCov[4*p + 1]);
            __builtin_nontemporal_store(c10, &outCov[4*p + 2]);
            __builtin_nontemporal_store(c11, &outCov[4*p + 3]);
            __builtin_nontemporal_store(mf, &outMask[p]);
        }
    }
}

extern "C" void kernel_launch(void* const* d_in, const int* in_sizes, int n_in,
                              void* d_out, int out_size, void* d_ws, size_t ws_size,
                              hipStream_t stream) {
    const float* sc  = (const float*)d_in[0];   // screen_coords [N,3]
    const float* pos = (const float*)d_in[1];   // pos3d         [N,3]
    const float* cov = (const float*)d_in[2];   // cov3d         [N,3,3]
    const float* w2m = (const float*)d_in[3];   // world2model   [4,4]
    const float* prj = (const float*)d_in[4];   // projection    [4,4]

    const int n = in_sizes[0] / 3;
    float* out = (float*)d_out;
    float* outPos  = out;                       // [N,3]
    float* outCov  = out + (size_t)3 * n;       // [N,2,2]
    float* outMask = out + (size_t)7 * n;       // [N]

    const int block = 256;
    long long blocksNeeded = ((long long)n + block - 1) / block;
    int grid = (int)(blocksNeeded < 4096 ? blocksNeeded : 4096);
    if (grid < 1) grid = 1;

    camera_project_kernel<<<grid, block, 0, stream>>>(
        sc, pos, cov, w2m, prj, outPos, outCov, outMask, n);
}